// DynamicSparseConv_53137335386843
// MI455X (gfx1250) — compile-verified
//
#include <hip/hip_runtime.h>
#include <hip/hip_bf16.h>

typedef __attribute__((ext_vector_type(2))) float v2f;
typedef __attribute__((ext_vector_type(4))) float v4f;
typedef __attribute__((ext_vector_type(8))) float v8f;

#define BB   32
#define CC   256
#define HH   56
#define WW   56
#define HWN  (HH * WW)      // 3136
#define CR   64             // C / 4
#define NL   2304           // 9 * C

// ---------------------------------------------------------------------------
// Kernel 1: global average pool over H*W for each (b,c). One block per (b,c).
// ---------------------------------------------------------------------------
__global__ void k_gap(const float* __restrict__ x, float* __restrict__ gap) {
  const int bc  = blockIdx.x;
  const int tid = threadIdx.x;
  const float* xp = x + (size_t)bc * HWN;

  float partial = 0.f;
  // 3136 floats = 784 float4s
  for (int i = tid; i < HWN / 4; i += 256) {
    v4f v = *(const v4f*)(xp + i * 4);
    partial += v.x + v.y + v.z + v.w;
  }

  __shared__ float red[256];
  red[tid] = partial;
  __syncthreads();
  for (int s = 128; s > 0; s >>= 1) {
    if (tid < s) red[tid] += red[tid + s];
    __syncthreads();
  }
  if (tid == 0) gap[bc] = red[0] * (1.0f / (float)HWN);
}

// ---------------------------------------------------------------------------
// Kernel 2: hid = gelu(gap @ w1^T + b1).  [32,256]x[256,64] via WMMA f32
// 16x16x4. One block, 8 waves = 2 M-tiles x 4 N-tiles.
// ---------------------------------------------------------------------------
__global__ void k_mlp1(const float* __restrict__ gap,
                       const float* __restrict__ w1,
                       const float* __restrict__ b1,
                       float* __restrict__ hid) {
  const int tid   = threadIdx.x;
  const int wave  = tid >> 5;
  const int lane  = tid & 31;
  const int row   = lane & 15;       // M row (A) / N col (B,C,D)
  const int half  = lane >> 4;       // 0: K={0,1} ; 1: K={2,3}

  const int mTile = wave & 1;        // 0..1  (M = 32)
  const int nTile = wave >> 1;       // 0..3  (N = 64)

  v8f acc = {};
  const float* arow = gap + (size_t)(mTile * 16 + row) * CC + 2 * half;
  const float* brow = w1  + (size_t)(nTile * 16 + row) * CC + 2 * half; // B[k][n]=w1[n][k]

  for (int k0 = 0; k0 < CC; k0 += 4) {
    v2f a = *(const v2f*)(arow + k0);
    v2f b = *(const v2f*)(brow + k0);
    acc = __builtin_amdgcn_wmma_f32_16x16x4_f32(
        /*neg_a=*/false, a, /*neg_b=*/false, b,
        /*c_mod=*/(short)0, acc, /*reuse_a=*/false, /*reuse_b=*/false);
  }

  const int n = nTile * 16 + row;
  const float bias = b1[n];
#pragma unroll
  for (int i = 0; i < 8; ++i) {
    const int m = mTile * 16 + i + 8 * half;
    float v = acc[i] + bias;
    // exact (erf) GELU
    v = 0.5f * v * (1.0f + erff(v * 0.70710678118654752f));
    hid[(size_t)m * CR + n] = v;
  }
}

// ---------------------------------------------------------------------------
// Kernel 3: logits = hid @ w2^T + b2.  [32,64]x[64,2304] via WMMA f32
// 16x16x4. 288 tiles = 36 blocks x 8 waves.
// ---------------------------------------------------------------------------
__global__ void k_mlp2(const float* __restrict__ hid,
                       const float* __restrict__ w2,
                       const float* __restrict__ b2,
                       float* __restrict__ logits) {
  const int tid  = threadIdx.x;
  const int wave = tid >> 5;
  const int lane = tid & 31;
  const int row  = lane & 15;
  const int half = lane >> 4;

  const int tile  = blockIdx.x * 8 + wave;   // 0..287
  const int mTile = tile & 1;                // M = 32 -> 2 tiles
  const int nTile = tile >> 1;               // N = 2304 -> 144 tiles

  v8f acc = {};
  const float* arow = hid + (size_t)(mTile * 16 + row) * CR + 2 * half;
  const float* brow = w2  + (size_t)(nTile * 16 + row) * CR + 2 * half; // B[k][n]=w2[n][k]

  for (int k0 = 0; k0 < CR; k0 += 4) {
    v2f a = *(const v2f*)(arow + k0);
    v2f b = *(const v2f*)(brow + k0);
    acc = __builtin_amdgcn_wmma_f32_16x16x4_f32(
        false, a, false, b, (short)0, acc, false, false);
  }

  const int n = nTile * 16 + row;
  const float bias = b2[n];
#pragma unroll
  for (int i = 0; i < 8; ++i) {
    const int m = mTile * 16 + i + 8 * half;
    logits[(size_t)m * NL + n] = acc[i] + bias;
  }
}

// ---------------------------------------------------------------------------
// Kernel 4: per-(b,c) softmax over 9 taps + dynamic 3x3 depthwise conv +
// residual. One block per (b,c) plane; plane staged in LDS with zero halo.
// ---------------------------------------------------------------------------
__global__ void k_dynconv(const float* __restrict__ x,
                          const float* __restrict__ logits,
                          float* __restrict__ y) {
  const int bc  = blockIdx.x;             // b*C + c
  const int b   = bc >> 8;                // / 256
  const int c   = bc & 255;
  const int tid = threadIdx.x;

  __shared__ float sw[9];
  __shared__ float t[58 * 58];            // halo-padded plane

  if (tid == 0) {
    const float* lp = logits + (size_t)b * NL + c * 9;
    float l[9], mx = lp[0];
#pragma unroll
    for (int k = 0; k < 9; ++k) { l[k] = lp[k]; mx = fmaxf(mx, l[k]); }
    float s = 0.f;
#pragma unroll
    for (int k = 0; k < 9; ++k) { l[k] = __expf(l[k] - mx); s += l[k]; }
    const float inv = 1.0f / s;
#pragma unroll
    for (int k = 0; k < 9; ++k) sw[k] = l[k] * inv;
  }

  // zero the whole tile (halo), then load interior
  for (int i = tid; i < 58 * 58; i += 256) t[i] = 0.f;
  __syncthreads();

  const float* xp = x + (size_t)bc * HWN;
  for (int g = tid; g < HWN / 4; g += 256) {
    const int r  = g / 14;                // 14 float4s per row
    const int cc = (g % 14) * 4;
    v4f v = *(const v4f*)(xp + r * WW + cc);
    float* dst = &t[(r + 1) * 58 + cc + 1];
    dst[0] = v.x; dst[1] = v.y; dst[2] = v.z; dst[3] = v.w;
  }
  __syncthreads();

  float w9[9];
#pragma unroll
  for (int k = 0; k < 9; ++k) w9[k] = sw[k];

  float* yp = y + (size_t)bc * HWN;
  for (int g = tid; g < HWN / 4; g += 256) {
    const int r  = g / 14;
    const int cc = (g % 14) * 4;
    float acc0 = t[(r + 1) * 58 + cc + 1];   // residual (center)
    float acc1 = t[(r + 1) * 58 + cc + 2];
    float acc2 = t[(r + 1) * 58 + cc + 3];
    float acc3 = t[(r + 1) * 58 + cc + 4];
#pragma unroll
    for (int kh = 0; kh < 3; ++kh) {
#pragma unroll
      for (int kw = 0; kw < 3; ++kw) {
        const float wv = w9[kh * 3 + kw];
        const float* sp = &t[(r + kh) * 58 + cc + kw];
        acc0 = fmaf(wv, sp[0], acc0);
        acc1 = fmaf(wv, sp[1], acc1);
        acc2 = fmaf(wv, sp[2], acc2);
        acc3 = fmaf(wv, sp[3], acc3);
      }
    }
    v4f out; out.x = acc0; out.y = acc1; out.z = acc2; out.w = acc3;
    *(v4f*)(yp + r * WW + cc) = out;
  }
}

// ---------------------------------------------------------------------------
extern "C" void kernel_launch(void* const* d_in, const int* in_sizes, int n_in,
                              void* d_out, int out_size, void* d_ws, size_t ws_size,
                              hipStream_t stream) {
  const float* x  = (const float*)d_in[0];   // [32,256,56,56]
  const float* w1 = (const float*)d_in[1];   // [64,256]
  const float* b1 = (const float*)d_in[2];   // [64]
  const float* w2 = (const float*)d_in[3];   // [2304,64]
  const float* b2 = (const float*)d_in[4];   // [2304]
  float* y = (float*)d_out;

  float* ws     = (float*)d_ws;
  float* gap    = ws;                 // 8192 floats
  float* hid    = ws + 8192;          // 2048 floats
  float* logits = ws + 10240;         // 73728 floats  (total ~328 KB)

  k_gap    <<<BB * CC, 256, 0, stream>>>(x, gap);
  k_mlp1   <<<1,       256, 0, stream>>>(gap, w1, b1, hid);
  k_mlp2   <<<36,      256, 0, stream>>>(hid, w2, b2, logits);
  k_dynconv<<<BB * CC, 256, 0, stream>>>(x, logits, y);
}